// GCN_audio_top_med_37434934952164
// MI455X (gfx1250) — compile-verified
//
#include <hip/hip_runtime.h>
#include <hip/hip_bf16.h>
#include <math.h>

typedef __attribute__((ext_vector_type(2))) float v2f;
typedef __attribute__((ext_vector_type(8))) float v8f;

#define CIN   1024
#define COUT  256
#define KNODE 16
#define HWPIX 196
#define NB    128
#define BNEPS 1e-5f

// ---- gfx1250 async global->LDS support (probe with fallback) ----
#if defined(__gfx1250__) && __has_builtin(__builtin_amdgcn_global_load_async_to_lds_b128)
#define HAVE_ASYNC_LDS 1
#else
#define HAVE_ASYNC_LDS 0
#endif

#if HAVE_ASYNC_LDS
// Param types per hipcc diagnostic: (int4 __device__*, int4 __shared__*, int, int)
typedef int v4i_gld __attribute__((vector_size(16)));
#define ASYNC_GPTR(p) ((__attribute__((address_space(1))) v4i_gld*)(uintptr_t)(p))
#define ASYNC_LPTR(p) ((__attribute__((address_space(3))) v4i_gld*)(unsigned int)(uintptr_t)(p))
#if __has_builtin(__builtin_amdgcn_s_wait_asynccnt)
#define WAIT_ASYNC(n) __builtin_amdgcn_s_wait_asynccnt(n)
#else
#define WAIT_ASYNC(n) asm volatile("s_wait_asynccnt %0" :: "n"(n) : "memory")
#endif
#endif

// ------------------------------------------------------------------
// Kernel A1: split-K partial channel sums. 8 channel-chunks per
// sample -> 1024 blocks of concurrent load streams (the 103MB image
// stream is the roofline term; this maximizes loads in flight).
// ------------------------------------------------------------------
__global__ void k_feat_partial(const float* __restrict__ images,
                               float* __restrict__ partial)
{
    const int n     = blockIdx.x >> 3;
    const int chunk = blockIdx.x & 7;
    const int t     = threadIdx.x;
    if (t < HWPIX) {
        const float* p = images + (size_t)n * CIN * HWPIX
                                + (size_t)chunk * 128 * HWPIX + t;
        float s = 0.f;
        #pragma unroll 8
        for (int c = 0; c < 128; ++c) {
            if ((c & 31) == 0 && (c + 32) < 128)
                __builtin_prefetch(p + (size_t)(c + 32) * HWPIX, 0, 0);
            s += p[(size_t)c * HWPIX];
        }
        partial[(size_t)blockIdx.x * HWPIX + t] = s;
    }
}

// ------------------------------------------------------------------
// Kernel A2: finish feat, exact stable descending argsort via rank
// counting, emit scene indices and rows/cols (floats in d_out tail).
// ------------------------------------------------------------------
__global__ void k_feat_scene(const float* __restrict__ partial,
                             int*  __restrict__ scene_g,
                             float* __restrict__ rows_out,
                             float* __restrict__ cols_out)
{
    __shared__ float feat[HWPIX];
    __shared__ int   scene_s[32];
    const int n = blockIdx.x;
    const int t = threadIdx.x;

    if (t < HWPIX) {
        float s = 0.f;
        #pragma unroll
        for (int c8 = 0; c8 < 8; ++c8)
            s += partial[(size_t)(n * 8 + c8) * HWPIX + t];
        feat[t] = s;
    }
    __syncthreads();

    if (t < HWPIX) {
        const float fi = feat[t];
        int rank = 0;
        #pragma unroll 4
        for (int j = 0; j < HWPIX; ++j) {
            const float fj = feat[j];
            rank += (fj > fi) || (fj == fi && j < t);   // stable desc sort
        }
        if (rank < 16)                      scene_s[rank] = t;           // topK
        else if (rank >= 89 && rank < 105)  scene_s[16 + rank - 89] = t; // avgK
    }
    __syncthreads();

    if (t < 32) {
        const int s = scene_s[t];
        scene_g[n * 32 + t]  = s;
        rows_out[n * 32 + t] = (float)(s / 14);
        cols_out[n * 32 + t] = (float)(s % 14);
    }
}

// ------------------------------------------------------------------
// Kernel B: conv1/conv2 GEMM  X = relu(nodes @ W^T + b)
// A-tiles read DIRECTLY from images via scene index (no staging).
// One wave -> one 16x16 fp32 tile via V_WMMA_F32_16X16X4_F32.
// ------------------------------------------------------------------
__global__ void k_conv_gemm(const float* __restrict__ images,
                            const int*  __restrict__ scene,
                            const float* __restrict__ w1,
                            const float* __restrict__ b1,
                            const float* __restrict__ w2,
                            const float* __restrict__ b2,
                            float* __restrict__ xtop,
                            float* __restrict__ xmed)
{
    const int wave = (blockIdx.x * blockDim.x + threadIdx.x) >> 5;  // 0..4095
    const int lane = threadIdx.x & 31;
    const int mat   = wave >> 11;          // 0 = top, 1 = med
    const int rem   = wave & 2047;
    const int mtile = rem >> 4;            // sample index 0..127
    const int ntile = rem & 15;            // 0..15

    const float* W  = mat ? w2 : w1;
    const float* Bv = mat ? b2 : b1;
    float*       X  = mat ? xmed : xtop;

    const int m    = lane & 15;
    const int koff = (lane >> 4) << 1;     // 0 or 2

    const int n  = mtile;
    const int hw = scene[n * 32 + mat * 16 + m];
    const float* abase = images + (size_t)n * CIN * HWPIX + hw;

    const int    ncol = ntile * 16 + (lane & 15);
    const float* bcol = W + (size_t)ncol * CIN;

    v8f acc = {};
    #pragma unroll 4
    for (int k0 = 0; k0 < CIN; k0 += 4) {
        v2f a, b;
        a.x = abase[(size_t)(k0 + koff)     * HWPIX];
        a.y = abase[(size_t)(k0 + koff + 1) * HWPIX];
        b.x = bcol[k0 + koff];
        b.y = bcol[k0 + koff + 1];
        acc = __builtin_amdgcn_wmma_f32_16x16x4_f32(
                  false, a, false, b, (short)0, acc, false, false);
    }

    const float bias = Bv[ncol];
    #pragma unroll
    for (int r = 0; r < 8; ++r) {
        const int row = mtile * 16 + r + ((lane >> 4) << 3);
        float v = acc[r] + bias;
        v = v > 0.f ? v : 0.f;
        X[(size_t)row * COUT + ncol] = v;
    }
}

// ------------------------------------------------------------------
// Kernel C: per-sample normalized adjacency L (16x16) in LDS, then
// y = L @ x with WMMA (A-operand from LDS).
// ------------------------------------------------------------------
__global__ void k_graph_mix(const int*  __restrict__ scene,
                            const float* __restrict__ xtop,
                            const float* __restrict__ xmed,
                            float* __restrict__ cat)
{
    __shared__ float Amat[2][16][16];
    __shared__ float Lm[2][16][16];
    __shared__ float dinv[2][16];

    const int n = blockIdx.x;
    const int t = threadIdx.x;
    const int i = t >> 4, j = t & 15;

    #pragma unroll
    for (int which = 0; which < 2; ++which) {
        const int si = scene[n * 32 + which * 16 + i];
        const int sj = scene[n * 32 + which * 16 + j];
        const float dr = (float)(si / 14) - (float)(sj / 14);
        const float dc = (float)(si % 14) - (float)(sj % 14);
        const float dist = sqrtf(dr * dr + dc * dc);
        Amat[which][i][j] = 1.0f / (1.0f + dist);
    }
    __syncthreads();
    if (t < 32) {
        const int w = t >> 4, ii = t & 15;
        float s = 0.f;
        #pragma unroll
        for (int jj = 0; jj < 16; ++jj) s += Amat[w][ii][jj];
        dinv[w][ii] = rsqrtf(s);
    }
    __syncthreads();
    #pragma unroll
    for (int which = 0; which < 2; ++which)
        Lm[which][i][j] = Amat[which][i][j] * dinv[which][i] * dinv[which][j];
    __syncthreads();

    const int wv = t >> 5, lane = t & 31;
    const int m = lane & 15;
    const int koff = (lane >> 4) << 1;

    for (int job = wv; job < 32; job += 8) {
        const int which = job >> 4;
        const int ntile = job & 15;
        const float* X  = which ? xmed : xtop;
        const int ncol  = ntile * 16 + (lane & 15);

        v8f acc = {};
        #pragma unroll
        for (int k0 = 0; k0 < 16; k0 += 4) {
            v2f a, b;
            a.x = Lm[which][m][k0 + koff];
            a.y = Lm[which][m][k0 + koff + 1];
            b.x = X[(size_t)(n * 16 + k0 + koff)     * COUT + ncol];
            b.y = X[(size_t)(n * 16 + k0 + koff + 1) * COUT + ncol];
            acc = __builtin_amdgcn_wmma_f32_16x16x4_f32(
                      false, a, false, b, (short)0, acc, false, false);
        }
        #pragma unroll
        for (int r = 0; r < 8; ++r) {
            const int row = r + ((lane >> 4) << 3);
            cat[(size_t)n * 8192 + which * 4096 + row * COUT + ncol] = acc[r];
        }
    }
}

// ------------------------------------------------------------------
// Kernel D: fp32 WMMA GEMM  Z = Hin(128,K) @ W(C,K)^T + bias.
// Block = 1 mtile x 8 ntiles (8 waves). The shared 16xK A-panel is
// staged into LDS in double-buffered K-chunks with async global->LDS
// (ASYNCcnt) when available; B (no reuse) streams from global.
// ------------------------------------------------------------------
#define CKF 512
#define CKP (CKF + 4)   // +4 floats row pad: A reads spread across LDS banks

__global__ void k_fc_gemm(const float* __restrict__ Hin,
                          const float* __restrict__ W,
                          const float* __restrict__ bias,
                          float* __restrict__ Z,
                          int K, int Cout)
{
    __shared__ float As[2][16][CKP];

    const int tid  = threadIdx.x;
    const int wv   = tid >> 5;
    const int lane = tid & 31;
    const int ngroups = Cout >> 7;            // (Cout/16)/8
    const int mtile = blockIdx.x / ngroups;
    const int group = blockIdx.x % ngroups;
    const int ntile = group * 8 + wv;

    const int m    = lane & 15;
    const int koff = (lane >> 4) << 1;
    const int ncol = ntile * 16 + (lane & 15);
    const float* bcol = W + (size_t)ncol * K;
    const int nch = K / CKF;

    // cooperative stage of 16 x CKF A-chunk kc into As[kc&1]
    auto stage = [&](int kc) {
        const int buf = kc & 1;
        #pragma unroll
        for (int it = 0; it < (16 * CKF / 4) / 256; ++it) {     // 8 iters
            const int g   = tid + it * 256;
            const int r   = g / (CKF / 4);
            const int kk4 = (g % (CKF / 4)) * 4;
            const float* gp = Hin + (size_t)(mtile * 16 + r) * K
                                  + (size_t)kc * CKF + kk4;
            float* lp = &As[buf][r][kk4];
#if HAVE_ASYNC_LDS
            __builtin_amdgcn_global_load_async_to_lds_b128(
                ASYNC_GPTR(gp), ASYNC_LPTR(lp), 0, 0);
#else
            *(float4*)lp = *(const float4*)gp;
#endif
        }
    };

    v8f acc = {};
    stage(0);
    for (int kc = 0; kc < nch; ++kc) {
        const int buf = kc & 1;
        if (kc + 1 < nch) {
            stage(kc + 1);                     // prefetch next chunk (other buf)
#if HAVE_ASYNC_LDS
            WAIT_ASYNC(8);                     // drain chunk kc (8 wave-ops), keep kc+1 in flight
#endif
        } else {
#if HAVE_ASYNC_LDS
            WAIT_ASYNC(0);
#endif
        }
        __syncthreads();                       // all waves' chunk kc visible

        const float* bc = bcol + (size_t)kc * CKF;
        #pragma unroll 4
        for (int k0 = 0; k0 < CKF; k0 += 4) {
            v2f a, b;
            a.x = As[buf][m][k0 + koff];
            a.y = As[buf][m][k0 + koff + 1];
            b.x = bc[k0 + koff];
            b.y = bc[k0 + koff + 1];
            acc = __builtin_amdgcn_wmma_f32_16x16x4_f32(
                      false, a, false, b, (short)0, acc, false, false);
        }
        __syncthreads();                       // buf reusable next round
    }

    const float bv = bias[ncol];
    #pragma unroll
    for (int r = 0; r < 8; ++r) {
        const int row = mtile * 16 + r + ((lane >> 4) << 3);
        Z[(size_t)row * Cout + ncol] = acc[r] + bv;
    }
}

// ------------------------------------------------------------------
// Kernel E: batch-norm over batch axis (128 rows), block per column.
// ------------------------------------------------------------------
__global__ void k_bn(const float* __restrict__ Z,
                     float* __restrict__ out,
                     const float* __restrict__ g,
                     const float* __restrict__ b,
                     int cols, int do_relu)
{
    __shared__ float red[128];
    __shared__ float mean_s, var_s;
    const int c = blockIdx.x;
    const int r = threadIdx.x;

    const float z = Z[(size_t)r * cols + c];
    red[r] = z;
    __syncthreads();
    for (int s = 64; s > 0; s >>= 1) { if (r < s) red[r] += red[r + s]; __syncthreads(); }
    if (r == 0) mean_s = red[0] * (1.0f / 128.0f);
    __syncthreads();
    const float d = z - mean_s;
    red[r] = d * d;
    __syncthreads();
    for (int s = 64; s > 0; s >>= 1) { if (r < s) red[r] += red[r + s]; __syncthreads(); }
    if (r == 0) var_s = red[0] * (1.0f / 128.0f);
    __syncthreads();

    float y = g[c] * d * rsqrtf(var_s + BNEPS) + b[c];
    if (do_relu) y = fmaxf(y, 0.0f);
    out[(size_t)r * cols + c] = y;
}

// ------------------------------------------------------------------
// Kernel F: tiny fc3 (128x7, K=512).
// ------------------------------------------------------------------
__global__ void k_fc3(const float* __restrict__ H2,
                      const float* __restrict__ W,
                      const float* __restrict__ bias,
                      float* __restrict__ Z3)
{
    const int id = blockIdx.x * blockDim.x + threadIdx.x;
    if (id >= NB * 7) return;
    const int n = id / 7, c = id % 7;
    const float* h = H2 + (size_t)n * 512;
    const float* w = W  + (size_t)c * 512;
    float s = 0.f;
    #pragma unroll 8
    for (int k = 0; k < 512; ++k) s += h[k] * w[k];
    Z3[id] = s + bias[c];
}

// ------------------------------------------------------------------
// Kernel G: row softmax over 7 classes.
// ------------------------------------------------------------------
__global__ void k_softmax(const float* __restrict__ Z3n,
                          float* __restrict__ out)
{
    const int n = threadIdx.x;
    const float* z = Z3n + n * 7;
    float mx = z[0];
    #pragma unroll
    for (int c = 1; c < 7; ++c) mx = fmaxf(mx, z[c]);
    float e[7], s = 0.f;
    #pragma unroll
    for (int c = 0; c < 7; ++c) { e[c] = expf(z[c] - mx); s += e[c]; }
    const float inv = 1.0f / s;
    #pragma unroll
    for (int c = 0; c < 7; ++c) out[n * 7 + c] = e[c] * inv;
}

// ------------------------------------------------------------------
// Workspace layout (bytes)
// ------------------------------------------------------------------
#define OFF_SCENE 0u                          // 128*32*4         = 16384
#define OFF_PART  16384u                      // 128*8*196*4      = 802816
#define OFF_XTOP  (OFF_PART + 802816u)        // 2048*256*4       = 2097152
#define OFF_XMED  (OFF_XTOP + 2097152u)
#define OFF_CAT   (OFF_XMED + 2097152u)       // 128*8192*4       = 4194304
#define OFF_Z1    (OFF_CAT + 4194304u)        // 128*1024*4       = 524288
#define OFF_H1    (OFF_Z1 + 524288u)
#define OFF_Z2    (OFF_H1 + 524288u)          // 128*512*4        = 262144
#define OFF_H2    (OFF_Z2 + 262144u)
#define OFF_Z3    (OFF_H2 + 262144u)          // 4096
#define OFF_Z3N   (OFF_Z3 + 4096u)

extern "C" void kernel_launch(void* const* d_in, const int* in_sizes, int n_in,
                              void* d_out, int out_size, void* d_ws, size_t ws_size,
                              hipStream_t stream)
{
    const float* images  = (const float*)d_in[0];
    const float* conv1_w = (const float*)d_in[1];
    const float* conv1_b = (const float*)d_in[2];
    const float* conv2_w = (const float*)d_in[3];
    const float* conv2_b = (const float*)d_in[4];
    const float* fc1_w   = (const float*)d_in[5];
    const float* fc1_b   = (const float*)d_in[6];
    const float* fc2_w   = (const float*)d_in[7];
    const float* fc2_b   = (const float*)d_in[8];
    const float* fc3_w   = (const float*)d_in[9];
    const float* fc3_b   = (const float*)d_in[10];
    const float* bn1_g   = (const float*)d_in[11];
    const float* bn1_b   = (const float*)d_in[12];
    const float* bn2_g   = (const float*)d_in[13];
    const float* bn2_b   = (const float*)d_in[14];
    const float* bn3_g   = (const float*)d_in[15];
    const float* bn3_b   = (const float*)d_in[16];

    char* ws = (char*)d_ws;
    int*   scene   = (int*)  (ws + OFF_SCENE);
    float* partial = (float*)(ws + OFF_PART);
    float* xtop    = (float*)(ws + OFF_XTOP);
    float* xmed    = (float*)(ws + OFF_XMED);
    float* cat     = (float*)(ws + OFF_CAT);
    float* Z1      = (float*)(ws + OFF_Z1);
    float* H1      = (float*)(ws + OFF_H1);
    float* Z2      = (float*)(ws + OFF_Z2);
    float* H2      = (float*)(ws + OFF_H2);
    float* Z3      = (float*)(ws + OFF_Z3);
    float* Z3n     = (float*)(ws + OFF_Z3N);

    float* probs    = (float*)d_out;             // 128*7
    float* rows_out = probs + NB * 7;            // 128*32
    float* cols_out = rows_out + NB * 32;        // 128*32

    // A: split-K channel sums (1024 concurrent blocks), then rank/scene
    k_feat_partial<<<NB * 8, 224, 0, stream>>>(images, partial);
    k_feat_scene<<<NB, 224, 0, stream>>>(partial, scene, rows_out, cols_out);

    // B: conv GEMMs (4096 waves = 512 blocks x 8 waves)
    k_conv_gemm<<<512, 256, 0, stream>>>(images, scene, conv1_w, conv1_b,
                                         conv2_w, conv2_b, xtop, xmed);

    // C: Laplacian build + graph mix -> cat (N, 8192)
    k_graph_mix<<<NB, 256, 0, stream>>>(scene, xtop, xmed, cat);

    // D: fc1 (8 mtiles x 8 ntile-groups = 64 blocks), async-LDS A staging
    k_fc_gemm<<<64, 256, 0, stream>>>(cat, fc1_w, fc1_b, Z1, 8192, 1024);
    k_bn<<<1024, 128, 0, stream>>>(Z1, H1, bn1_g, bn1_b, 1024, 1);

    // fc2 (8 x 4 = 32 blocks)
    k_fc_gemm<<<32, 256, 0, stream>>>(H1, fc2_w, fc2_b, Z2, 1024, 512);
    k_bn<<<512, 128, 0, stream>>>(Z2, H2, bn2_g, bn2_b, 512, 1);

    // fc3 + bn + softmax
    k_fc3<<<4, 256, 0, stream>>>(H2, fc3_w, fc3_b, Z3);
    k_bn<<<7, 128, 0, stream>>>(Z3, Z3n, bn3_g, bn3_b, 7, 0);
    k_softmax<<<1, 128, 0, stream>>>(Z3n, probs);
}